// InputLayer_74594991997073
// MI455X (gfx1250) — compile-verified
//
#include <hip/hip_runtime.h>
#include <stdint.h>

// ---------------------------------------------------------------------------
// SCN InputLayer mode=3: dedupe (sorted) coordinate keys, scatter-add feature
// rows of duplicate points into the unique voxel row (row = rank of key).
//
// Key space: S=64 -> key in [0, 2^24). Strategy:
//   1) bitmap[2^24 bits] presence via atomicOr
//   2) hierarchical exclusive prefix sum of per-word popcounts ->
//      packed[w] = { bitmap word, exclusive popcount offset }  (one b64/lookup)
//   3) rank(key) = packed.y + popc(packed.x & belowMask)
//   4) wave-per-point scatter: 4-deep ring of async global->LDS copies of the
//      128B feature row (GLOBAL_LOAD_ASYNC_TO_LDS_B32, ASYNCcnt-gated),
//      overlapping rank lookup; then coalesced global_atomic_add_f32.
// ---------------------------------------------------------------------------

#define SDIM            64
#define KEYSPACE        (SDIM * SDIM * SDIM * SDIM)   // 16,777,216
#define NWORDS          (KEYSPACE / 32)               // 524,288 u32 words
#define WORDS_PER_CHUNK 256
#define NCHUNKS         (NWORDS / WORDS_PER_CHUNK)    // 2,048
#define PIPE_DEPTH      4

// ---------------------------- zero fill ------------------------------------
__global__ void k_zero_u128(uint4* __restrict__ p, long long n4) {
  long long i = (long long)blockIdx.x * blockDim.x + threadIdx.x;
  long long stride = (long long)gridDim.x * blockDim.x;
  uint4 z = make_uint4(0u, 0u, 0u, 0u);
  for (; i < n4; i += stride) p[i] = z;
}

__global__ void k_zero_u32(uint32_t* __restrict__ p, int n) {
  int i = blockIdx.x * blockDim.x + threadIdx.x;
  if (i < n) p[i] = 0u;
}

// ---------------------------- bitmap build ---------------------------------
__global__ void k_build_bitmap(const int* __restrict__ coords,
                               uint32_t* __restrict__ bitmap, int n) {
  int i = blockIdx.x * blockDim.x + threadIdx.x;
  if (i >= n) return;
  int4 c = ((const int4*)coords)[i];
  uint32_t key = (((uint32_t)c.x * 64u + (uint32_t)c.y) * 64u +
                  (uint32_t)c.z) * 64u + (uint32_t)c.w;
  atomicOr(&bitmap[key >> 5], 1u << (key & 31u));
}

// ------------------ hierarchical prefix sum of popcounts -------------------
__global__ void k_chunk_sums(const uint32_t* __restrict__ bitmap,
                             uint32_t* __restrict__ chunkSum) {
  __shared__ uint32_t sh[WORDS_PER_CHUNK];
  const int t = threadIdx.x;
  sh[t] = (uint32_t)__popc(bitmap[blockIdx.x * WORDS_PER_CHUNK + t]);
  __syncthreads();
  for (int s = WORDS_PER_CHUNK / 2; s > 0; s >>= 1) {
    if (t < s) sh[t] += sh[t + s];
    __syncthreads();
  }
  if (t == 0) chunkSum[blockIdx.x] = sh[0];
}

__global__ void k_scan_chunks(const uint32_t* __restrict__ chunkSum,
                              uint32_t* __restrict__ chunkOff) {
  __shared__ uint32_t sh[256];
  const int t = threadIdx.x;
  const int PER = NCHUNKS / 256;  // 8
  uint32_t loc[NCHUNKS / 256];
  uint32_t run = 0;
#pragma unroll
  for (int j = 0; j < PER; ++j) {
    uint32_t v = chunkSum[t * PER + j];
    loc[j] = run;
    run += v;
  }
  sh[t] = run;
  __syncthreads();
  for (int off = 1; off < 256; off <<= 1) {
    uint32_t v = (t >= off) ? sh[t - off] : 0u;
    __syncthreads();
    sh[t] += v;
    __syncthreads();
  }
  uint32_t base = sh[t] - run;  // exclusive across threads
#pragma unroll
  for (int j = 0; j < PER; ++j) chunkOff[t * PER + j] = base + loc[j];
}

// Emits packed[w] = { bitmap word, exclusive rank offset of word } so the
// scatter does ONE b64 load per rank lookup instead of two b32s.
__global__ void k_word_offsets(const uint32_t* __restrict__ bitmap,
                               const uint32_t* __restrict__ chunkOff,
                               uint2* __restrict__ packed) {
  __shared__ uint32_t sh[WORDS_PER_CHUNK];
  const int t = threadIdx.x;
  const int w = blockIdx.x * WORDS_PER_CHUNK + t;
  const uint32_t bits = bitmap[w];
  const uint32_t pc = (uint32_t)__popc(bits);
  sh[t] = pc;
  __syncthreads();
  for (int off = 1; off < WORDS_PER_CHUNK; off <<= 1) {
    uint32_t v = (t >= off) ? sh[t - off] : 0u;
    __syncthreads();
    sh[t] += v;
    __syncthreads();
  }
  packed[w] = make_uint2(bits, chunkOff[blockIdx.x] + sh[t] - pc);
}

// ------------------------------ scatter ------------------------------------
// One wave (32 lanes) per point; lane c handles channel c. Feature rows flow
// through a 4-deep ring of CDNA5 async global->LDS copies (ASYNCcnt completes
// in order, so waiting <= outstanding-1 guarantees the head slot is ready).
__global__ void k_scatter_c32(const int* __restrict__ coords,
                              const float* __restrict__ feat,
                              const uint2* __restrict__ packed,
                              float* __restrict__ out, int n) {
  __shared__ float buf[8][PIPE_DEPTH][32];  // 8 waves/block, 4 ring slots
  const int lane = threadIdx.x & 31;
  const int wl   = threadIdx.x >> 5;
  const int wid  = (int)((blockIdx.x * blockDim.x + threadIdx.x) >> 5);
  const int nw   = (int)((gridDim.x * blockDim.x) >> 5);

  // Prologue: fill the ring (points wid + k*nw, k = 0..3).
#pragma unroll
  for (int k = 0; k < PIPE_DEPTH; ++k) {
    int q = wid + k * nw;
    if (q < n) {
      uint32_t ldsa = (uint32_t)(uintptr_t)&buf[wl][k][lane];
      uint64_t ga   = (uint64_t)(uintptr_t)(feat + (size_t)q * 32 + lane);
      asm volatile("global_load_async_to_lds_b32 %0, %1, off"
                   :: "v"(ldsa), "v"(ga) : "memory");
    }
  }

  int slot = 0;
  for (int p = wid; p < n; p += nw) {
    // Rank lookup overlaps the in-flight async copies. Index is wave-uniform;
    // readfirstlane lets the compiler use scalar loads (K$ broadcast).
    const int ps = __builtin_amdgcn_readfirstlane(p);
    int4 c = ((const int4*)coords)[ps];
    uint32_t key = (((uint32_t)c.x * 64u + (uint32_t)c.y) * 64u +
                    (uint32_t)c.z) * 64u + (uint32_t)c.w;
    uint2 pw = packed[key >> 5];
    uint32_t rank = pw.y +
        (uint32_t)__popc(pw.x & ((1u << (key & 31u)) - 1u));

    // Outstanding (issued, unconsumed) async loads cover p..p+3*nw clipped.
    const int r = 1 + (int)(p + nw < n) + (int)(p + 2 * nw < n) +
                  (int)(p + 3 * nw < n);
    switch (r) {  // wave-uniform
      case 4:  asm volatile("s_wait_asynccnt 3" ::: "memory"); break;
      case 3:  asm volatile("s_wait_asynccnt 2" ::: "memory"); break;
      case 2:  asm volatile("s_wait_asynccnt 1" ::: "memory"); break;
      default: asm volatile("s_wait_asynccnt 0" ::: "memory"); break;
    }

    float v = buf[wl][slot][lane];
    // atomic consumes v -> compiler's s_wait_dscnt lands before it, so the
    // LDS read is complete before we recycle this slot below (no WAR race).
    atomicAdd(out + (size_t)rank * 32 + lane, v);

    int q = p + PIPE_DEPTH * nw;
    if (q < n) {
      uint32_t ldsa = (uint32_t)(uintptr_t)&buf[wl][slot][lane];
      uint64_t ga   = (uint64_t)(uintptr_t)(feat + (size_t)q * 32 + lane);
      asm volatile("global_load_async_to_lds_b32 %0, %1, off"
                   :: "v"(ldsa), "v"(ga) : "memory");
    }
    slot = (slot + 1) & (PIPE_DEPTH - 1);
  }
}

// Fallback for C != 32 (reference fixes C=32; kept for robustness).
__global__ void k_scatter_generic(const int* __restrict__ coords,
                                  const float* __restrict__ feat,
                                  const uint2* __restrict__ packed,
                                  float* __restrict__ out, int n, int C) {
  int i = blockIdx.x * blockDim.x + threadIdx.x;
  if (i >= n) return;
  int4 c = ((const int4*)coords)[i];
  uint32_t key = (((uint32_t)c.x * 64u + (uint32_t)c.y) * 64u +
                  (uint32_t)c.z) * 64u + (uint32_t)c.w;
  uint2 pw = packed[key >> 5];
  uint32_t rank = pw.y + (uint32_t)__popc(pw.x & ((1u << (key & 31u)) - 1u));
  for (int ch = 0; ch < C; ++ch)
    atomicAdd(out + (size_t)rank * C + ch, feat[(size_t)i * C + ch]);
}

// ------------------------------ launcher -----------------------------------
extern "C" void kernel_launch(void* const* d_in, const int* in_sizes, int n_in,
                              void* d_out, int out_size, void* d_ws, size_t ws_size,
                              hipStream_t stream) {
  const int*   coords = (const int*)d_in[0];
  const float* feat   = (const float*)d_in[1];
  float*       out    = (float*)d_out;

  const int n = in_sizes[0] / 4;
  const int C = (n > 0) ? (in_sizes[1] / n) : 0;

  uint8_t* ws = (uint8_t*)d_ws;
  uint32_t* bitmap   = (uint32_t*)ws;                          // 2 MB @ 0
  uint2*    packed   = (uint2*)(ws + (size_t)NWORDS * 4);      // 4 MB @ 2 MB
  uint32_t* chunkSum = (uint32_t*)(ws + (size_t)NWORDS * 12);  // 8 KB @ 6 MB
  uint32_t* chunkOff = chunkSum + NCHUNKS;                     // 8 KB

  // 1) zero bitmap (poisoned ws) and output (poisoned out)
  k_zero_u128<<<256, 256, 0, stream>>>((uint4*)bitmap, NWORDS / 4);
  long long n4 = out_size / 4;
  if (n4 > 0) k_zero_u128<<<2048, 256, 0, stream>>>((uint4*)out, n4);
  int tail = out_size & 3;
  if (tail) k_zero_u32<<<1, 64, 0, stream>>>((uint32_t*)out + (size_t)n4 * 4, tail);

  if (n <= 0 || C <= 0) return;

  // 2) presence bitmap
  k_build_bitmap<<<(n + 255) / 256, 256, 0, stream>>>(coords, bitmap, n);

  // 3) exclusive prefix of popcounts -> packed {bits, rank offset}
  k_chunk_sums  <<<NCHUNKS, WORDS_PER_CHUNK, 0, stream>>>(bitmap, chunkSum);
  k_scan_chunks <<<1, 256, 0, stream>>>(chunkSum, chunkOff);
  k_word_offsets<<<NCHUNKS, WORDS_PER_CHUNK, 0, stream>>>(bitmap, chunkOff, packed);

  // 4) scatter-add features into ranked rows
  if (C == 32) {
    k_scatter_c32<<<2048, 256, 0, stream>>>(coords, feat, packed, out, n);
  } else {
    k_scatter_generic<<<(n + 255) / 256, 256, 0, stream>>>(coords, feat, packed,
                                                           out, n, C);
  }
}